// RP_11081015624366
// MI455X (gfx1250) — compile-verified
//
#include <hip/hip_runtime.h>
#include <hip/hip_bf16.h>
#include <math.h>

// Problem constants (from reference)
#define N_ROWS   256      // N*C = 64*4
#define ROW_LEN  512      // L
#define OUT_SZ   65       // OUTPUT_SIZE
#define TILES    5        // ceil(65/16)
#define THRESH   0.2f
#define EPS_VAL  1e-6f

typedef __attribute__((ext_vector_type(2))) float v2f;
typedef __attribute__((ext_vector_type(8))) float v8f;

// One wave32 per (row, tileI, tileJ) job.
// Wave recomputes row stats (cheap, L2-resident), builds the 16x16 pairwise
// difference tile on the matrix unit (v_wmma_f32_16x16x4_f32), thresholds,
// and stores the guarded 65x65 region.
//
// Transposed mapping (abs makes the plot symmetric in the difference sign):
//   D[m][n] = xn[tj*16+m] - xn[ti*16+n]
// so a lane (n = lane&15) owns output ROW i = ti*16+n, and its 8 accumulator
// VGPRs are 8 CONSECUTIVE output columns j = tj*16 + mofs + r. This makes the
// epilogue: one wave-uniform branch on tj + one exec-mask per wave, with
// contiguous per-lane stores (no per-element exec toggling).
__global__ __launch_bounds__(256) void RP_recurrence_wmma_kernel(
    const float* __restrict__ x, float* __restrict__ out) {
  const int lane = threadIdx.x & 31;
  const int waveInBlock = threadIdx.x >> 5;
  const int job = blockIdx.x * (blockDim.x >> 5) + waveInBlock;
  // job = row*25 + ti*5 + tj   (6400 jobs total, divides grid exactly)
  const int row = job / (TILES * TILES);
  const int t   = job % (TILES * TILES);
  const int ti  = t / TILES;
  const int tj  = t % TILES;

  const float* xr = x + (size_t)row * ROW_LEN;

  // ---- pass 1: first/last nonzero index over the row (wave32 reduction) ----
  int mn = 0x7fffffff;
  int mx = -1;
  for (int i = lane; i < ROW_LEN; i += 32) {
    float v = xr[i];
    if (v != 0.0f) {
      mn = (i < mn) ? i : mn;
      mx = (i > mx) ? i : mx;
    }
  }
  #pragma unroll
  for (int off = 16; off > 0; off >>= 1) {
    int omn = __shfl_xor(mn, off, 32);
    int omx = __shfl_xor(mx, off, 32);
    mn = (omn < mn) ? omn : mn;
    mx = (omx > mx) ? omx : mx;
  }

  // ---- pass 2: masked sum / sumsq (wave32 reduction) ----
  float sum = 0.0f, sumsq = 0.0f;
  for (int i = lane; i < ROW_LEN; i += 32) {
    float v = xr[i];
    bool in = (mx >= 0) && (i >= mn) && (i <= mx);
    float vm = in ? v : 0.0f;
    sum   += vm;
    sumsq += vm * vm;
  }
  #pragma unroll
  for (int off = 16; off > 0; off >>= 1) {
    sum   += __shfl_xor(sum, off, 32);
    sumsq += __shfl_xor(sumsq, off, 32);
  }

  const float cnt  = (mx >= 0) ? (float)(mx - mn + 1) : 1.0f;
  const float mean = sum / cnt;
  float var = sumsq / cnt - mean * mean;
  var = fmaxf(var, 0.0f);
  const float invd = 1.0f / fmaxf(sqrtf(var), EPS_VAL);

  // ---- build WMMA operands ----
  // A (16x4, MxK), ISA layout: lanes 0-15 -> M=lane, VGPR0=K0, VGPR1=K1;
  //                            lanes 16-31 -> K2,K3 (all zero here)
  //   A[:,0] = xn over the tj tile (output COLUMNS), A[:,1] = 1
  // B (4x16, KxN): lanes 0-15 -> N=lane, VGPR0=K0 row, VGPR1=K1 row;
  //                lanes 16-31 -> K2,K3 rows (zero)
  //   B[0,:] = 1, B[1,:] = -xn over the ti tile (output ROWS)
  // => D[m][n] = xn[tj*16+m] - xn[ti*16+n]
  const int p   = lane & 15;
  const int gjA = tj * 16 + p;  // feeds A (output columns), 0..79 < ROW_LEN
  const int giB = ti * 16 + p;  // feeds B (output rows)

  float xnj = 0.0f, xni = 0.0f;
  {
    float vj = xr[gjA];
    float vi = xr[giB];
    bool inj = (mx >= 0) && (gjA >= mn) && (gjA <= mx);
    bool ini = (mx >= 0) && (giB >= mn) && (giB <= mx);
    xnj = inj ? (vj - mean) * invd : 0.0f;
    xni = ini ? (vi - mean) * invd : 0.0f;
  }

  const bool lo = (lane < 16);
  v2f a, b;
  a.x = lo ? xnj  : 0.0f;   // A[:,0] = xn_cols | A[:,2] = 0
  a.y = lo ? 1.0f : 0.0f;   // A[:,1] = 1       | A[:,3] = 0
  b.x = lo ? 1.0f : 0.0f;   // B[0,:] = 1       | B[2,:] = 0
  b.y = lo ? -xni : 0.0f;   // B[1,:] = -xn_rows| B[3,:] = 0

  v8f c = {};
  // D = A x B + 0  -> pairwise differences, exact f32
  v8f d = __builtin_amdgcn_wmma_f32_16x16x4_f32(
      /*neg_a=*/false, a, /*neg_b=*/false, b,
      /*c_mod=*/(short)0, c, /*reuse_a=*/false, /*reuse_b=*/false);

  // ---- threshold + store ----
  // D layout: VGPR r, lane L -> element (M = r + (L<16?0:8), N = L%16)
  //   output row i = ti*16 + N   (fixed per lane)
  //   output col j = tj*16 + mofs + r   (8 consecutive columns per lane)
  const int n     = lane & 15;
  const int i     = ti * 16 + n;
  const int mofs  = lo ? 0 : 8;
  const int jbase = tj * 16 + mofs;

  if (i < OUT_SZ) {                       // single exec-mask (only trims ti==4)
    float* p0 = out + (size_t)row * OUT_SZ * OUT_SZ + (size_t)i * OUT_SZ + jbase;
    if (tj < TILES - 1) {                 // wave-uniform: interior tiles, 8 cols all valid
      #pragma unroll
      for (int r = 0; r < 8; ++r) {
        p0[r] = (fabsf(d[r]) < THRESH) ? 1.0f : 0.0f;
      }
    } else if (lo) {                      // tj==4: only j==64 exists (lanes<16, r==0)
      p0[0] = (fabsf(d[0]) < THRESH) ? 1.0f : 0.0f;
    }
  }
}

extern "C" void kernel_launch(void* const* d_in, const int* in_sizes, int n_in,
                              void* d_out, int out_size, void* d_ws, size_t ws_size,
                              hipStream_t stream) {
  (void)in_sizes; (void)n_in; (void)out_size; (void)d_ws; (void)ws_size;
  const float* x = (const float*)d_in[0];   // [64,4,512] f32, row-major
  float* out = (float*)d_out;               // [64,4,65,65] f32

  // 256 rows * 25 tiles = 6400 wave-jobs; 8 waves per 256-thread block -> 800 blocks
  const int jobs = N_ROWS * TILES * TILES;
  const int wavesPerBlock = 256 / 32;
  const int blocks = jobs / wavesPerBlock;  // divides exactly
  RP_recurrence_wmma_kernel<<<blocks, 256, 0, stream>>>(x, out);
}